// ImportanceSampler_15281493639381
// MI455X (gfx1250) — compile-verified
//
#include <hip/hip_runtime.h>
#include <hip/hip_bf16.h>
#include <stdint.h>

// Problem sizes (fixed by the reference)
#define NRAYS 131072
#define NS 64          // coarse samples per ray
#define NI 128         // importance samples per ray
#define NALL 192       // NS + NI

typedef __attribute__((ext_vector_type(2))) float v2f;
typedef __attribute__((ext_vector_type(8))) float v8f;

// Per-ray LDS scratch. One wave32 owns one ray.
struct __align__(16) RaySmem {
    float zv[64];     // z_vals
    float w[64];      // weights
    float zmid[64];   // 63 bins used
    float cdf[64];    // 63 entries used
    float zsamp[NI];  // importance samples (monotone)
    float zall[NALL]; // merged z's
    float pts[NALL * 3]; // staged WMMA output, 2304B (16B aligned: 576 floats before it)
};

__global__ __launch_bounds__(256) void importance_sampler_kernel(
    const float* __restrict__ rays_o,
    const float* __restrict__ rays_d,
    const float* __restrict__ z_vals,
    const float* __restrict__ weights,
    float* __restrict__ pts_out,   // [NRAYS,192,3]
    float* __restrict__ zall_out,  // [NRAYS,192]
    float* __restrict__ zs_out)    // [NRAYS,128]
{
    __shared__ RaySmem sm[8];
    const int lane = threadIdx.x & 31;
    const int wv   = threadIdx.x >> 5;
    const int ray  = blockIdx.x * 8 + wv;
    RaySmem& s = sm[wv];

    // ---------------- Phase 0: async-copy z_vals + weights -> LDS ----------
    // One global_load_async_to_lds_b128 per wave: lanes 0-15 bring 256B of
    // z_vals, lanes 16-31 bring 256B of weights. Tracked by ASYNCcnt.
    {
        const float* g = (lane < 16)
            ? (z_vals  + (size_t)ray * NS + lane * 4)
            : (weights + (size_t)ray * NS + (lane - 16) * 4);
        float* ldst = (lane < 16) ? &s.zv[lane * 4] : &s.w[(lane - 16) * 4];
        uint32_t loff = (uint32_t)(uintptr_t)ldst; // low 32b of generic ptr == LDS addr
        asm volatile("global_load_async_to_lds_b128 %0, %1, off"
                     :: "v"(loff), "v"(g) : "memory");
        asm volatile("s_wait_asynccnt 0x0" ::: "memory");
    }
    __syncthreads();

    // ---------------- Phase 1: bins (z_mid) and CDF -------------------------
    for (int i = lane; i < 63; i += 32)
        s.zmid[i] = 0.5f * (s.zv[i] + s.zv[i + 1]);

    // w has 62 interior weights: w[i] = weights[i+1] + 1e-5, i = 0..61.
    // Each lane owns a pair; 5-step wave32 inclusive shuffle scan.
    float e0 = (2 * lane     < 62) ? (s.w[2 * lane + 1] + 1e-5f) : 0.0f;
    float e1 = (2 * lane + 1 < 62) ? (s.w[2 * lane + 2] + 1e-5f) : 0.0f;
    float pairsum = e0 + e1;
    float incl = pairsum;
    #pragma unroll
    for (int d = 1; d < 32; d <<= 1) {
        float t = __shfl_up(incl, d, 32);
        if (lane >= d) incl += t;
    }
    float total = __shfl(incl, 31, 32);           // sum of all 62 weights
    float excl  = incl - pairsum;
    if (lane == 0) s.cdf[0] = 0.0f;
    if (lane <= 30) {                              // writes cdf[1..62]
        s.cdf[2 * lane + 1] = (excl + e0) / total;
        s.cdf[2 * lane + 2] = incl / total;
    }
    __syncthreads();

    // ---------------- Phase 2: inverse-CDF sampling (128 samples) ----------
    for (int t = 0; t < 4; ++t) {
        int sj = lane + 32 * t;
        float u = (sj == 127) ? 1.0f : sj * (1.0f / 127.0f);
        // searchsorted(cdf, u, side='right') over 63 entries
        int lo = 0, hi = 63;
        while (lo < hi) { int m = (lo + hi) >> 1; if (s.cdf[m] <= u) lo = m + 1; else hi = m; }
        int below = lo - 1; if (below < 0) below = 0;
        int above = lo;     if (above > 62) above = 62;
        float cb = s.cdf[below], ca = s.cdf[above];
        float bb = s.zmid[below], ba = s.zmid[above];
        float den = ca - cb; if (den < 1e-5f) den = 1.0f;
        float zs = bb + ((u - cb) / den) * (ba - bb);
        s.zsamp[sj] = zs;
        zs_out[(size_t)ray * NI + sj] = zs;        // coalesced
    }
    __syncthreads();

    // ---------------- Phase 3: merge (both lists are sorted) ----------------
    // Stable merge-path: A=z_vals placed by lower_bound into z_samples,
    // B=z_samples placed by upper_bound into z_vals -> bijective positions.
    for (int t = 0; t < 2; ++t) {
        int i = lane + 32 * t;
        float v = s.zv[i];
        int lo = 0, hi = NI;
        while (lo < hi) { int m = (lo + hi) >> 1; if (s.zsamp[m] < v) lo = m + 1; else hi = m; }
        s.zall[i + lo] = v;
    }
    for (int t = 0; t < 4; ++t) {
        int j = lane + 32 * t;
        float v = s.zsamp[j];
        int lo = 0, hi = NS;
        while (lo < hi) { int m = (lo + hi) >> 1; if (s.zv[m] <= v) lo = m + 1; else hi = m; }
        s.zall[j + lo] = v;
    }
    __syncthreads();
    for (int t = 0; t < 6; ++t) {
        int k = lane + 32 * t;
        zall_out[(size_t)ray * NALL + k] = s.zall[k];  // coalesced
    }

    // ---------------- Phase 4: pts via V_WMMA_F32_16X16X4_F32 --------------
    // D(16x16) = A(16x4) x B(4x16):
    //   A[m,0] = z_all[chunk*16+m], A[m,1] = 1, A[m,2..3] = 0
    //   B[0,n] = rays_d[n], B[1,n] = rays_o[n] (n<3), rows 2..3 = 0
    // => D[m,n] = z*d[n] + o[n] = pts. ISA 32-bit layouts:
    //   A: VGPR0 = {lanes0-15:K0, lanes16-31:K2}, VGPR1 = {K1, K3}
    //   B: VGPR0 = {lanes0-15:rowK0, lanes16-31:rowK2}, VGPR1 = {rowK1, rowK3}
    float dcomp = 0.0f, ocomp = 0.0f;
    if (lane < 3) {
        dcomp = rays_d[(size_t)ray * 3 + lane];
        ocomp = rays_o[(size_t)ray * 3 + lane];
    }
    v2f B;
    B.x = dcomp;   // lanes 0-2 carry d, everything else (incl. rows K2/K3) = 0
    B.y = ocomp;   // lanes 0-2 carry o
    #pragma unroll 1
    for (int c = 0; c < 12; ++c) {
        v2f A;
        A.x = (lane < 16) ? s.zall[c * 16 + lane] : 0.0f;  // K0 col = z, K2 col = 0
        A.y = (lane < 16) ? 1.0f : 0.0f;                   // K1 col = 1, K3 col = 0
        v8f C = {};
        v8f D = __builtin_amdgcn_wmma_f32_16x16x4_f32(
            /*neg_a=*/false, A, /*neg_b=*/false, B,
            /*c_mod=*/(short)0, C, /*reuse_a=*/false, /*reuse_b=*/false);
        // D layout: VGPR v -> lanes0-15: (M=v, N=lane); lanes16-31: (M=v+8, N=lane-16)
        int M0 = (lane < 16) ? 0 : 8;
        int N  = (lane < 16) ? lane : lane - 16;
        if (N < 3) {
            #pragma unroll
            for (int v = 0; v < 8; ++v)
                s.pts[(c * 16 + M0 + v) * 3 + N] = D[v];
        }
    }
    __syncthreads();

    // Coalesced float2 stores of the staged 576 floats (9 per lane).
    const float2* p2 = (const float2*)s.pts;
    float2* o2 = (float2*)(pts_out + (size_t)ray * (NALL * 3));
    #pragma unroll
    for (int t = 0; t < 9; ++t)
        o2[lane + 32 * t] = p2[lane + 32 * t];
}

extern "C" void kernel_launch(void* const* d_in, const int* in_sizes, int n_in,
                              void* d_out, int out_size, void* d_ws, size_t ws_size,
                              hipStream_t stream) {
    const float* rays_o  = (const float*)d_in[0];
    const float* rays_d  = (const float*)d_in[1];
    const float* z_vals  = (const float*)d_in[2];
    const float* weights = (const float*)d_in[3];

    float* out      = (float*)d_out;
    float* pts_out  = out;                                      // [N,192,3]
    float* zall_out = pts_out + (size_t)NRAYS * NALL * 3;       // [N,192]
    float* zs_out   = zall_out + (size_t)NRAYS * NALL;          // [N,128]

    dim3 grid(NRAYS / 8), block(256);
    importance_sampler_kernel<<<grid, block, 0, stream>>>(
        rays_o, rays_d, z_vals, weights, pts_out, zall_out, zs_out);
}